// RNN_9560597201506
// MI455X (gfx1250) — compile-verified
//
#include <hip/hip_runtime.h>
#include <math.h>

// Problem constants (match reference)
#define SEQ_L   512
#define BATCH   64
#define DIM     512
#define LATENT  1024

typedef __attribute__((ext_vector_type(2))) float        v2f;
typedef __attribute__((ext_vector_type(8))) float        v8f;
typedef __attribute__((ext_vector_type(4))) unsigned int u32x4;
typedef __attribute__((ext_vector_type(8))) int          i32x8;
typedef __attribute__((ext_vector_type(4))) int          i32x4;

#if defined(__has_builtin)
#if __has_builtin(__builtin_amdgcn_tensor_load_to_lds)
#define HAVE_TDM 1
#endif
#endif

// V_WMMA_F32_16X16X4_F32 wrapper (fragment layout per ISA 7.12.2, wave32):
//   A : lane l -> { A[l&15][2*(l>>4)+j] }   B : lane l -> { W[l&15][2*(l>>4)+j] }
//   C/D VGPR r -> M = r + 8*(l>>4), N = l&15
__device__ __forceinline__ v8f mm4(v2f a, v2f b, v8f c) {
    return __builtin_amdgcn_wmma_f32_16x16x4_f32(false, a, false, b, 0, c,
                                                 false, false);
}

// ---------------------------------------------------------------------------
// Kernel 1: xi = x @ Wi^T + Wi_b  ->  d_out  (layout [L*B, LATENT]).
// Register-tiled: each wave computes a 32x64 output block (2 A-frags x
// 4 B-frags -> 8 WMMAs per 6 b64 loads per k-step).
// Tiles: 1024 row-groups x 16 col-groups = 16384 waves = 2048 blocks.
// ---------------------------------------------------------------------------
__global__ __launch_bounds__(256) void rnn_xi_gemm(
    const float* __restrict__ x,     // [L*B, DIM]
    const float* __restrict__ Wi,    // [LATENT, DIM]
    const float* __restrict__ bi,    // [LATENT]
    float* __restrict__ out)         // [L*B, LATENT]
{
    const int lane = threadIdx.x & 31;
    const int gw   = blockIdx.x * 8 + (threadIdx.x >> 5);
    const int tm   = gw >> 4;        // 32-row group (0..1023)
    const int tn   = gw & 15;        // 64-col group (0..15)
    const int half = lane >> 4;
    const int sub  = lane & 15;

    const float* a0 = x + (size_t)(tm * 32 + sub) * DIM + 2 * half;
    const float* a1 = a0 + 16 * DIM;
    const float* b0 = Wi + (size_t)(tn * 64 + sub) * DIM + 2 * half;
    const float* b1 = b0 + 16 * DIM;
    const float* b2 = b0 + 32 * DIM;
    const float* b3 = b0 + 48 * DIM;

    v8f c[2][4];
#pragma unroll
    for (int i = 0; i < 2; ++i)
#pragma unroll
        for (int j = 0; j < 4; ++j)
            c[i][j] = (v8f){0.f, 0.f, 0.f, 0.f, 0.f, 0.f, 0.f, 0.f};

#pragma unroll 2
    for (int k = 0; k < DIM; k += 4) {
        v2f A0 = *(const v2f*)(a0 + k);
        v2f A1 = *(const v2f*)(a1 + k);
        v2f B0 = *(const v2f*)(b0 + k);
        v2f B1 = *(const v2f*)(b1 + k);
        v2f B2 = *(const v2f*)(b2 + k);
        v2f B3 = *(const v2f*)(b3 + k);
        c[0][0] = mm4(A0, B0, c[0][0]);
        c[0][1] = mm4(A0, B1, c[0][1]);
        c[0][2] = mm4(A0, B2, c[0][2]);
        c[0][3] = mm4(A0, B3, c[0][3]);
        c[1][0] = mm4(A1, B0, c[1][0]);
        c[1][1] = mm4(A1, B1, c[1][1]);
        c[1][2] = mm4(A1, B2, c[1][2]);
        c[1][3] = mm4(A1, B3, c[1][3]);
    }

#pragma unroll
    for (int j = 0; j < 4; ++j) {
        const int col  = tn * 64 + j * 16 + sub;
        const float bias = bi[col];
#pragma unroll
        for (int r = 0; r < 8; ++r) {
            const int row0 = tm * 32 + r + 8 * half;
            out[(size_t)row0 * LATENT + col]        = c[0][j][r] + bias;
            out[(size_t)(row0 + 16) * LATENT + col] = c[1][j][r] + bias;
        }
    }
}

// ---------------------------------------------------------------------------
// Kernel 2: one recurrence step, out_t = tanh(xi + h_prev @ Wh^T + Wh_b).
// One block (8 waves) per latent-column tile tn (64 blocks). The 16x1024
// Wh strip for tn is DMA'd once into LDS by the Tensor Data Mover (padded
// 2 DWORDs per 256 to break bank conflicts). Waves split K 8 ways; each wave
// computes all 4 batch tiles from one LDS B-fragment (4 WMMAs per B-load).
// Partials are reduced through LDS (region aliased over the Wh copy).
//
// LDS: 16 rows * 1024 dw + (16384/256)*2 pad dw = 16512 dw = 66048 B.
// ---------------------------------------------------------------------------
#define STEP_LDS_DW   16512
#define STEP_LDS_BYTES (STEP_LDS_DW * 4)

__global__ __launch_bounds__(256) void rnn_step(
    const float* __restrict__ hprev, // [B, LATENT]
    const float* __restrict__ Wh,    // [LATENT, LATENT]
    const float* __restrict__ bh,    // [LATENT]
    float* __restrict__ out_t)       // [B, LATENT]: holds xi_t, overwritten h_t
{
    extern __shared__ float smem[];
    const int tid  = threadIdx.x;
    const int wid  = tid >> 5;
    const int lane = tid & 31;
    const int tn   = blockIdx.x;     // latent-col tile (0..63)
    const int half = lane >> 4;
    const int sub  = lane & 15;

    // ---- Stage Wh[tn*16 .. tn*16+15][0..1023] into LDS ---------------------
#if defined(HAVE_TDM)
    if (wid == 0) {
        const unsigned lds_base =
            (unsigned)(uintptr_t)(void*)smem;   // low 32 bits = LDS byte offset
        const unsigned long long ga =
            (unsigned long long)(uintptr_t)(Wh + (size_t)tn * 16 * LATENT);
        u32x4 g0;
        g0[0] = 1u;                                   // count = 1 valid D#
        g0[1] = lds_base;                             // lds_addr
        g0[2] = (unsigned)ga;                         // global_addr[31:0]
        g0[3] = (unsigned)((ga >> 32) & 0x01FFFFFFu)  // global_addr[56:32]
                | (2u << 30);                         // type = 2 (image)
        i32x8 g1;
        g1[0] = (int)((2u << 16)     // data_size = 4B
                      | (1u << 20)   // pad_enable
                      | (7u << 22)   // pad_interval: every 256 DWORDs
                      | (1u << 25)); // pad_amount: 2 DWORDs
        g1[1] = (int)((LATENT & 0xFFFFu) << 16);      // tensor_dim0[15:0]
        g1[2] = (int)((LATENT >> 16) | (16u << 16));  // dim0[31:16] | dim1[15:0]
        g1[3] = (int)(1024u << 16);                   // dim1[31:16] | tile_dim0
        g1[4] = 16;                                   // tile_dim1 | tile_dim2=0
        g1[5] = LATENT;                               // tensor_dim0_stride lo
        g1[6] = 0;
        g1[7] = 0;
        i32x4 gz = {0, 0, 0, 0};
#if defined(__clang_major__) && __clang_major__ >= 23
        i32x8 gz8 = {0, 0, 0, 0, 0, 0, 0, 0};
        __builtin_amdgcn_tensor_load_to_lds(g0, g1, gz, gz, gz8, 0);
#else
        __builtin_amdgcn_tensor_load_to_lds(g0, g1, gz, gz, 0);
#endif
        __builtin_amdgcn_s_wait_tensorcnt(0);
    }
#else
    // Cooperative fallback with identical padded layout.
    {
        const float* src = Wh + (size_t)tn * 16 * LATENT;
        for (int dw = tid; dw < 16 * LATENT; dw += 256)
            smem[dw + ((dw >> 8) << 1)] = src[dw];
    }
#endif
    __syncthreads();

    // ---- K-sliced GEMM: this wave handles K in [kbase, kbase+128) ----------
    const int kbase = wid * 128;
    const float* ap0 = hprev + (size_t)(0 * 16 + sub) * LATENT + kbase + 2 * half;
    const float* ap1 = ap0 + 16 * LATENT;
    const float* ap2 = ap0 + 32 * LATENT;
    const float* ap3 = ap0 + 48 * LATENT;

    v8f acc[4];
#pragma unroll
    for (int i = 0; i < 4; ++i)
        acc[i] = (v8f){0.f, 0.f, 0.f, 0.f, 0.f, 0.f, 0.f, 0.f};

#pragma unroll 4
    for (int kk = 0; kk < 128; kk += 4) {
        const int off = sub * LATENT + kbase + kk + 2 * half;   // even
        v2f B  = *(const v2f*)(smem + off + ((off >> 8) << 1)); // padded LDS
        v2f A0 = *(const v2f*)(ap0 + kk);
        v2f A1 = *(const v2f*)(ap1 + kk);
        v2f A2 = *(const v2f*)(ap2 + kk);
        v2f A3 = *(const v2f*)(ap3 + kk);
        acc[0] = mm4(A0, B, acc[0]);
        acc[1] = mm4(A1, B, acc[1]);
        acc[2] = mm4(A2, B, acc[2]);
        acc[3] = mm4(A3, B, acc[3]);
    }
    __syncthreads();   // all waves done reading the Wh copy

    // ---- Write partials (aliased over Wh region): [w][tm][lane][r] ---------
#pragma unroll
    for (int tm = 0; tm < 4; ++tm) {
        float* p = smem + (size_t)(((wid * 4 + tm) * 32 + lane) * 8);
#pragma unroll
        for (int r = 0; r < 8; ++r) p[r] = acc[tm][r];
    }
    __syncthreads();

    // ---- Reduce 8 K-slices, add xi + bias, tanh, store ---------------------
    // 64 rows x 16 cols = 1024 outputs; 4 per thread.
#pragma unroll
    for (int i = 0; i < 4; ++i) {
        const int o    = tid + i * 256;
        const int col  = o & 15;
        const int row  = o >> 4;              // 0..63
        const int tm   = row >> 4;
        const int m16  = row & 15;
        const int r    = m16 & 7;
        const int ln   = (m16 >> 3) * 16 + col;
        float s = 0.f;
#pragma unroll
        for (int w = 0; w < 8; ++w)
            s += smem[((w * 4 + tm) * 32 + ln) * 8 + r];
        const int cg = tn * 16 + col;
        const size_t idx = (size_t)row * LATENT + cg;
        out_t[idx] = tanhf(s + out_t[idx] + bh[cg]);
    }
}

// ---------------------------------------------------------------------------
// Launch: one big xi GEMM, then 512 stream-ordered step kernels.
// ---------------------------------------------------------------------------
extern "C" void kernel_launch(void* const* d_in, const int* in_sizes, int n_in,
                              void* d_out, int out_size, void* d_ws,
                              size_t ws_size, hipStream_t stream) {
    const float* x  = (const float*)d_in[0];  // [L,B,DIM]
    const float* h0 = (const float*)d_in[1];  // [B,LATENT] (zeros)
    const float* Wi = (const float*)d_in[2];  // [LATENT,DIM]
    const float* bi = (const float*)d_in[3];  // [LATENT]
    const float* Wh = (const float*)d_in[4];  // [LATENT,LATENT]
    const float* bh = (const float*)d_in[5];  // [LATENT]
    float* out = (float*)d_out;               // [L,B,LATENT]

    // xi for all timesteps -> d_out
    const int xi_blocks = (SEQ_L * BATCH / 32) * (LATENT / 64) / 8; // 2048
    rnn_xi_gemm<<<xi_blocks, 256, 0, stream>>>(x, Wi, bi, out);

    // sequential recurrence: one block per latent-column tile
    for (int t = 0; t < SEQ_L; ++t) {
        const float* hp =
            (t == 0) ? h0 : out + (size_t)(t - 1) * BATCH * LATENT;
        rnn_step<<<LATENT / 16, 256, STEP_LDS_BYTES, stream>>>(
            hp, Wh, bh, out + (size_t)t * BATCH * LATENT);
    }
}